// RWKV_LAYER_53618371723281
// MI455X (gfx1250) — compile-verified
//
#include <hip/hip_runtime.h>
#include <math.h>

#define L_N 24
#define C_N 1024
#define F_N 4096
#define LN_EPS 1e-5f

typedef float v2f __attribute__((ext_vector_type(2)));
typedef float v8f __attribute__((ext_vector_type(8)));

// ---------------------------------------------------------------------------
// Wave-level GEMV over 16 rows using V_WMMA_F32_16X16X4_F32.
// A tile (16x4 f32): lane m (0..15) holds row m, K = {k,k+1}; lanes 16..31
// hold row m, K = {k+2,k+3}  -> one 8-byte load per lane per step.
// B tile (4x16 f32): we broadcast x[k+r] across all 16 columns, so every
// column of D equals the same partial GEMV; D VGPR j holds row j (lanes 0-15)
// and row j+8 (lanes 16-31).
// ---------------------------------------------------------------------------
__device__ __forceinline__ v8f wmma_rows16(const float* __restrict__ W, int ldw,
                                           const float* __restrict__ xs,
                                           int kbeg, int kend) {
  const int lane = threadIdx.x & 31;
  const int m    = lane & 15;
  const int sel  = lane >> 4;                     // 0 or 1
  const float* wp = W + (size_t)m * ldw + kbeg + 2 * sel;
  v8f acc = {0.f, 0.f, 0.f, 0.f, 0.f, 0.f, 0.f, 0.f};
#pragma unroll 4
  for (int k = kbeg; k < kend; k += 4) {
    v2f a = *(const v2f*)wp;                      // global_load_b64
    v2f b;
    b.x = xs[k + sel];                            // B rows 0/1 (all columns)
    b.y = xs[k + 2 + sel];                        // B rows 2/3 (all columns)
    acc = __builtin_amdgcn_wmma_f32_16x16x4_f32(false, a, false, b,
                                                (short)0, acc, false, false);
    wp += 4;
  }
  return acc;
}

// Store per-wave 16-row partials to LDS: lane 0 holds rows 0..7 in acc[0..7],
// lane 16 holds rows 8..15.
__device__ __forceinline__ void wave_dump(v8f acc, float (*red)[16], int wv) {
  const int lane = threadIdx.x & 31;
  if ((lane & 15) == 0) {
    const int rb = (lane >> 4) * 8;
#pragma unroll
    for (int j = 0; j < 8; ++j) red[wv][rb + j] = acc[j];
  }
}

__device__ __forceinline__ float block_sum(float v, float* sred) {
  const int t = threadIdx.x;
  sred[t] = v;
  __syncthreads();
  for (int s = 128; s > 0; s >>= 1) {
    if (t < s) sred[t] += sred[t + s];
    __syncthreads();
  }
  float r = sred[0];
  __syncthreads();
  return r;
}

// ---------------------------------------------------------------------------
// Kernel 1: ln1 (recomputed per block) + k/v/r GEMVs.  grid = 3*64 blocks.
// ---------------------------------------------------------------------------
__global__ void k_kvr(const float* __restrict__ xin, const float* __restrict__ state,
                      const float* __restrict__ ln1w, const float* __restrict__ ln1b,
                      const float* __restrict__ kktk, const float* __restrict__ vvtv,
                      const float* __restrict__ rrtr, const float* __restrict__ key,
                      float* __restrict__ kvr_out, float* __restrict__ sa_out, int layer) {
  __shared__ float xs[C_N];
  __shared__ float red[8][16];
  __shared__ float sred[256];
  __shared__ float s_m, s_r;
  const int t = threadIdx.x, bid = blockIdx.x;
  const int section = bid >> 6, tile = bid & 63;

  float lsum = 0.f, lsq = 0.f;
  for (int c = t; c < C_N; c += 256) { float v = xin[c]; lsum += v; lsq += v * v; }
  float tot  = block_sum(lsum, sred);
  float tot2 = block_sum(lsq, sred);
  if (t == 0) {
    float mm = tot / C_N;
    s_m = mm;
    s_r = rsqrtf(tot2 / C_N - mm * mm + LN_EPS);
  }
  __syncthreads();
  const float m = s_m, rstd = s_r;

  const float* mix = (section == 0) ? kktk : (section == 1) ? vvtv : rrtr;
  const float* sa  = state + (size_t)layer * C_N;          // state[0][layer]
  const float* w1  = ln1w + (size_t)layer * C_N;
  const float* b1  = ln1b + (size_t)layer * C_N;
  const float* mx  = mix  + (size_t)layer * C_N;
  for (int c = t; c < C_N; c += 256) {
    float xy = (xin[c] - m) * rstd * w1[c] + b1[c];
    xs[c] = xy + mx[c] * sa[c];
    if (bid == 0) sa_out[c] = xy;                          // new statea = xy
  }
  __syncthreads();

  const float* W = key + ((size_t)layer * 3 + section) * C_N * C_N +
                   (size_t)(tile * 16) * C_N;
  const int wv = t >> 5;
  v8f acc = wmma_rows16(W, C_N, xs, wv * 128, wv * 128 + 128);
  wave_dump(acc, red, wv);
  __syncthreads();
  if (t < 16) {
    float s = 0.f;
#pragma unroll
    for (int w2 = 0; w2 < 8; ++w2) s += red[w2][t];
    kvr_out[section * C_N + tile * 16 + t] = s;
  }
}

// ---------------------------------------------------------------------------
// Kernel 2: wkv elementwise (per-block, block 0 writes sb/sc) + outputv GEMV.
// grid = 64 blocks.
// ---------------------------------------------------------------------------
__global__ void k_wkv_out(const float* __restrict__ xin, const float* __restrict__ state,
                          const float* __restrict__ tfirst, const float* __restrict__ tdecay,
                          const float* __restrict__ kvr, const float* __restrict__ outv,
                          float* __restrict__ sxx_out, float* __restrict__ sb_out,
                          float* __restrict__ sc_out, int layer) {
  __shared__ float xs[C_N];
  __shared__ float red[8][16];
  const int t = threadIdx.x, bid = blockIdx.x;
  const float* sb = state + ((size_t)1 * L_N + layer) * C_N;
  const float* sc = state + ((size_t)2 * L_N + layer) * C_N;
  const float* tf = tfirst + (size_t)layer * C_N;
  const float* td = tdecay + (size_t)layer * C_N;
  for (int c = t; c < C_N; c += 256) {
    float k = kvr[c], v = kvr[C_N + c], r = kvr[2 * C_N + c];
    float sbv = sb[c], scv = sc[c];
    float etfk = expf(tf[c] + k);
    float er   = expf(r);
    xs[c] = (sbv + etfk * v) / (scv * er + etfk * er + scv + etfk);
    if (bid == 0) {
      float ed = expf(td[c]);
      float ek = expf(k);
      sb_out[c] = sbv * ed + ek * v;
      sc_out[c] = scv * ed + ek;
    }
  }
  __syncthreads();

  const float* W = outv + (size_t)layer * C_N * C_N + (size_t)(bid * 16) * C_N;
  const int wv = t >> 5;
  v8f acc = wmma_rows16(W, C_N, xs, wv * 128, wv * 128 + 128);
  wave_dump(acc, red, wv);
  __syncthreads();
  if (t < 16) {
    float s = 0.f;
#pragma unroll
    for (int w2 = 0; w2 < 8; ++w2) s += red[w2][t];
    int row = bid * 16 + t;
    sxx_out[row] = xin[row] + s;
  }
}

// ---------------------------------------------------------------------------
// Kernel 3: ln2 (recomputed per block) + key_ffn GEMV (blocks 0..255) and
// receptance_ffn GEMV (blocks 256..319).  Block 256 writes stated (=xx).
// ---------------------------------------------------------------------------
__global__ void k_ffn_kr(const float* __restrict__ sxx, const float* __restrict__ state,
                         const float* __restrict__ ln2w, const float* __restrict__ ln2b,
                         const float* __restrict__ tmk, const float* __restrict__ tmr,
                         const float* __restrict__ kffn, const float* __restrict__ rffn,
                         float* __restrict__ kf_out, float* __restrict__ rf_out,
                         float* __restrict__ sd_out, int layer) {
  __shared__ float xs[C_N];
  __shared__ float red[8][16];
  __shared__ float sred[256];
  __shared__ float s_m, s_r;
  const int t = threadIdx.x, bid = blockIdx.x;
  const int sec  = (bid >= 256);
  const int tile = sec ? (bid - 256) : bid;

  float lsum = 0.f, lsq = 0.f;
  for (int c = t; c < C_N; c += 256) { float v = sxx[c]; lsum += v; lsq += v * v; }
  float tot  = block_sum(lsum, sred);
  float tot2 = block_sum(lsq, sred);
  if (t == 0) {
    float mm = tot / C_N;
    s_m = mm;
    s_r = rsqrtf(tot2 / C_N - mm * mm + LN_EPS);
  }
  __syncthreads();
  const float m = s_m, rstd = s_r;

  const float* sd = state + ((size_t)3 * L_N + layer) * C_N;
  const float* w2 = ln2w + (size_t)layer * C_N;
  const float* b2 = ln2b + (size_t)layer * C_N;
  const float* tm = (sec ? tmr : tmk) + (size_t)layer * C_N;
  for (int c = t; c < C_N; c += 256) {
    float xx = (sxx[c] - m) * rstd * w2[c] + b2[c];
    xs[c] = xx + tm[c] * sd[c];
    if (bid == 256) sd_out[c] = xx;                        // new stated = xx
  }
  __syncthreads();

  const float* W = sec ? (rffn + (size_t)layer * C_N * C_N + (size_t)(tile * 16) * C_N)
                       : (kffn + (size_t)layer * F_N * C_N + (size_t)(tile * 16) * C_N);
  const int wv = t >> 5;
  v8f acc = wmma_rows16(W, C_N, xs, wv * 128, wv * 128 + 128);
  wave_dump(acc, red, wv);
  __syncthreads();
  if (t < 16) {
    float s = 0.f;
#pragma unroll
    for (int w2i = 0; w2i < 8; ++w2i) s += red[w2i][t];
    int row = tile * 16 + t;
    if (sec) {
      rf_out[row] = expf(s);                               // rffn pre-negated
    } else {
      float rl = fmaxf(s, 0.f);
      kf_out[row] = rl * rl;
    }
  }
}

// ---------------------------------------------------------------------------
// Kernel 4: value_ffn GEMV over K=4096 + final combine.  grid = 64 blocks.
// ---------------------------------------------------------------------------
__global__ void k_vffn(const float* __restrict__ sxx, const float* __restrict__ kf,
                       const float* __restrict__ rf, const float* __restrict__ vffn,
                       float* __restrict__ xout, int layer) {
  __shared__ float xs[F_N];
  __shared__ float red[8][16];
  const int t = threadIdx.x, bid = blockIdx.x;
  for (int c = t; c < F_N; c += 256) xs[c] = kf[c];
  __syncthreads();

  const float* W = vffn + (size_t)layer * C_N * F_N + (size_t)(bid * 16) * F_N;
  const int wv = t >> 5;
  v8f acc = wmma_rows16(W, F_N, xs, wv * 512, wv * 512 + 512);
  wave_dump(acc, red, wv);
  __syncthreads();
  if (t < 16) {
    float s = 0.f;
#pragma unroll
    for (int w2 = 0; w2 < 8; ++w2) s += red[w2][t];
    int row = bid * 16 + t;
    xout[row] = sxx[row] + s / (rf[row] + 1.f);
  }
}

// ---------------------------------------------------------------------------
extern "C" void kernel_launch(void* const* d_in, const int* in_sizes, int n_in,
                              void* d_out, int out_size, void* d_ws, size_t ws_size,
                              hipStream_t stream) {
  const float* x0    = (const float*)d_in[0];
  const float* state = (const float*)d_in[1];
  const float* ln1w  = (const float*)d_in[2];
  const float* ln1b  = (const float*)d_in[3];
  const float* ln2w  = (const float*)d_in[4];
  const float* ln2b  = (const float*)d_in[5];
  const float* tdec  = (const float*)d_in[6];
  const float* tfst  = (const float*)d_in[7];
  const float* kktk  = (const float*)d_in[8];
  const float* vvtv  = (const float*)d_in[9];
  const float* rrtr  = (const float*)d_in[10];
  const float* key   = (const float*)d_in[11];
  const float* outv  = (const float*)d_in[12];
  const float* tmk   = (const float*)d_in[13];
  const float* tmr   = (const float*)d_in[14];
  const float* kffn  = (const float*)d_in[15];
  const float* rffn  = (const float*)d_in[16];
  const float* vffn  = (const float*)d_in[17];

  float* out = (float*)d_out;
  float* ws  = (float*)d_ws;
  float* ws_x   = ws;                  // [C]
  float* ws_kvr = ws + C_N;            // [3C]
  float* ws_sxx = ws + 4 * C_N;        // [C]
  float* ws_kf  = ws + 5 * C_N;        // [F]
  float* ws_rf  = ws + 9 * C_N;        // [C]

  float* out_x  = out;                          // [C]
  float* out_sa = out + C_N;                    // [L,C]
  float* out_sb = out + C_N + 1 * L_N * C_N;
  float* out_sc = out + C_N + 2 * L_N * C_N;
  float* out_sd = out + C_N + 3 * L_N * C_N;

  for (int layer = 0; layer < L_N; ++layer) {
    const float* xin = (layer == 0) ? x0 : ws_x;
    k_kvr<<<192, 256, 0, stream>>>(xin, state, ln1w, ln1b, kktk, vvtv, rrtr, key,
                                   ws_kvr, out_sa + (size_t)layer * C_N, layer);
    k_wkv_out<<<64, 256, 0, stream>>>(xin, state, tfst, tdec, ws_kvr, outv, ws_sxx,
                                      out_sb + (size_t)layer * C_N,
                                      out_sc + (size_t)layer * C_N, layer);
    k_ffn_kr<<<320, 256, 0, stream>>>(ws_sxx, state, ln2w, ln2b, tmk, tmr, kffn, rffn,
                                      ws_kf, ws_rf, out_sd + (size_t)layer * C_N, layer);
    float* xo = (layer == L_N - 1) ? out_x : ws_x;
    k_vffn<<<64, 256, 0, stream>>>(ws_sxx, ws_kf, ws_rf, vffn, xo, layer);
  }
}